// KNeighborsClassifierModel_55843164783337
// MI455X (gfx1250) — compile-verified
//
#include <hip/hip_runtime.h>
#include <math.h>

typedef __attribute__((ext_vector_type(2))) float v2f;
typedef __attribute__((ext_vector_type(8))) float v8f;

#define KNN_K 16
#define NCLS  10
#define QTILE 16     // queries per block
#define CTILE 64     // train columns per slab (4 waves x 16)
#define DIM   128
#define NCHUNK (DIM/4)   // 32 K-chunks of 4

__device__ __forceinline__ void load_slab(const float* __restrict__ rowp, int koff,
                                          v2f (&b)[NCHUNK]) {
#pragma unroll
    for (int kc = 0; kc < NCHUNK; ++kc)
        b[kc] = *(const v2f*)(rowp + kc * 4 + koff);
}

__device__ __forceinline__ void compute_slab(const v2f (&a)[NCHUNK], const v2f (&b)[NCHUNK],
                                             const float (&mx2)[8], int hi,
                                             float* __restrict__ dst) {
    v8f accA = {0.f,0.f,0.f,0.f,0.f,0.f,0.f,0.f};
    v8f accB = {0.f,0.f,0.f,0.f,0.f,0.f,0.f,0.f};
    float ts = 0.f;
#pragma unroll
    for (int kc = 0; kc < NCHUNK / 2; ++kc) {
        const v2f b0 = b[2*kc], b1 = b[2*kc + 1];
        ts += b0.x * b0.x + b0.y * b0.y;
        ts += b1.x * b1.x + b1.y * b1.y;
        accA = __builtin_amdgcn_wmma_f32_16x16x4_f32(
                 false, a[2*kc],     false, b0, (short)0, accA, false, false);
        accB = __builtin_amdgcn_wmma_f32_16x16x4_f32(
                 false, a[2*kc + 1], false, b1, (short)0, accB, false, false);
    }
    // complete column norm: lane n holds k%4 in {0,1}, lane n+16 holds {2,3}
    float t2 = ts + __shfl_xor(ts, 16, 32);
#pragma unroll
    for (int r = 0; r < 8; ++r)
        dst[r + hi * 8] = mx2[r] + t2 - 2.0f * (accA[r] + accB[r]);
}

__global__ __launch_bounds__(128)
void knn_wmma_kernel(const float* __restrict__ x,
                     const float* __restrict__ train,
                     const int*   __restrict__ labels,
                     float* __restrict__ out,
                     int N, int M)
{
    __shared__ float xs[QTILE][DIM];       // 8 KB: x tile
    __shared__ float sx2[QTILE];           // query row norms
    __shared__ float dtile[CTILE][17];     // d^2 tile, padded stride 17
    __shared__ float md[128 * KNN_K];      // partial top-k d^2
    __shared__ int   mi[128 * KNN_K];      // partial top-k indices

    const int t    = threadIdx.x;
    const int lane = t & 31;
    const int wv   = t >> 5;
    const int hi   = (lane >= 16) ? 1 : 0;
    const int ln   = lane & 15;
    const int koff = hi * 2;
    const int qbase = blockIdx.x * QTILE;

    // ---- load x tile to LDS (float4 vectorized) ----
    {
        const float4* xsrc = (const float4*)x;
        float4* xdst = (float4*)(&xs[0][0]);
        for (int i = t; i < QTILE * (DIM/4); i += 128) {
            int row = i / (DIM/4);
            int q = qbase + row;
            int src = (q < N ? q : 0) * (DIM/4) + (i % (DIM/4));
            xdst[i] = xsrc[src];
        }
    }
    __syncthreads();
    if (t < QTILE) {
        float s = 0.f;
        for (int k = 0; k < DIM; ++k) s += xs[t][k] * xs[t][k];
        sx2[t] = s;
    }
    __syncthreads();

    // ---- hoist slab-invariant operands into registers ----
    v2f areg[NCHUNK];                       // A operands (x rows), fixed for all slabs
#pragma unroll
    for (int kc = 0; kc < NCHUNK; ++kc)
        areg[kc] = *(const v2f*)(&xs[ln][kc * 4 + koff]);
    float myx2[8];                          // row norms for the 8 C rows this lane owns
#pragma unroll
    for (int r = 0; r < 8; ++r)
        myx2[r] = sx2[r + hi * 8];

    // register-resident local top-16 on d^2
    float bd[KNN_K]; int bi[KNN_K];
#pragma unroll
    for (int i = 0; i < KNN_K; ++i) { bd[i] = INFINITY; bi[i] = 0x7FFFFFFF; }
    float worst = INFINITY;

    const int mrow = t & 15;   // query row owned in phase B
    const int csub = t >> 4;   // column subset 0..7
    const int jcol = wv * 16 + ln;          // owned column within a slab (loader role)
    float* dst = &dtile[jcol][0];

    // top-k scan of one slab's d^2 tile (8 columns owned by this thread)
    auto topk_scan = [&](int j0) {
#pragma unroll
        for (int s = 0; s < 8; ++s) {
            int col = s * 8 + csub;
            int j = j0 + col;
            if (j < M) {
                float d2 = fmaxf(dtile[col][mrow], 0.0f);
                if (d2 < worst) {
                    int wip = 0; float wvv = bd[0];
#pragma unroll
                    for (int i = 1; i < KNN_K; ++i) { if (bd[i] > wvv) { wvv = bd[i]; wip = i; } }
#pragma unroll
                    for (int i = 0; i < KNN_K; ++i) { if (i == wip) { bd[i] = d2; bi[i] = j; } }
                    wvv = bd[0];
#pragma unroll
                    for (int i = 1; i < KNN_K; ++i) wvv = fmaxf(wvv, bd[i]);
                    worst = wvv;
                }
            }
        }
    };

    const int nslab = (M + CTILE - 1) / CTILE;
    const int Mm1 = M - 1;

    // ---- software-pipelined main loop: double-buffered B in registers ----
    v2f breg0[NCHUNK], breg1[NCHUNK];
    {
        int j = jcol; if (j > Mm1) j = Mm1;
        load_slab(train + (size_t)j * DIM, koff, breg0);
    }
    for (int sl = 0; sl < nslab; sl += 2) {
        {   // issue loads for slab sl+1 before consuming slab sl
            int j = (sl + 1) * CTILE + jcol; if (j > Mm1) j = Mm1;
            load_slab(train + (size_t)j * DIM, koff, breg1);
        }
        compute_slab(areg, breg0, myx2, hi, dst);
        __syncthreads();
        topk_scan(sl * CTILE);
        __syncthreads();

        {   // issue loads for slab sl+2 before consuming slab sl+1
            int j = (sl + 2) * CTILE + jcol; if (j > Mm1) j = Mm1;
            load_slab(train + (size_t)j * DIM, koff, breg0);
        }
        compute_slab(areg, breg1, myx2, hi, dst);   // dummy tail slab: filtered by j<M below
        __syncthreads();
        topk_scan((sl + 1) * CTILE);
        __syncthreads();
    }

    // ---- dump partial lists ----
#pragma unroll
    for (int i = 0; i < KNN_K; ++i) { md[t * KNN_K + i] = bd[i]; mi[t * KNN_K + i] = bi[i]; }
    __syncthreads();

    // ---- merge 8x16 -> 16, weight, vote: one thread per query ----
    if (t < QTILE) {
        int q = qbase + t;
        if (q < N) {
            float fd[KNN_K]; int fi[KNN_K];
            for (int sel = 0; sel < KNN_K; ++sel) {
                float bdist = INFINITY; int bidx = 0x7FFFFFFF; int bpos = 0;
                for (int c = 0; c < 8; ++c) {
                    int tt = t + 16 * c;
                    for (int k = 0; k < KNN_K; ++k) {
                        int p = tt * KNN_K + k;
                        float d = md[p]; int id = mi[p];
                        if (d < bdist || (d == bdist && id < bidx)) {
                            bdist = d; bidx = id; bpos = p;
                        }
                    }
                }
                fd[sel] = bdist; fi[sel] = bidx;
                md[bpos] = INFINITY;
            }
            // distances + exact-match (inf weight) handling
            int hasz = 0;
            for (int i = 0; i < KNN_K; ++i) {
                float d = sqrtf(fmaxf(fd[i], 0.0f));
                fd[i] = d;
                if (d == 0.0f) hasz = 1;
            }
            float proba[NCLS];
            for (int c = 0; c < NCLS; ++c) proba[c] = 0.f;
            for (int i = 0; i < KNN_K; ++i) {
                float w = hasz ? ((fd[i] == 0.0f) ? 1.0f : 0.0f) : (1.0f / fd[i]);
                int lab = labels[fi[i]];
                for (int c = 0; c < NCLS; ++c) proba[c] += (lab == c) ? w : 0.0f;
            }
            float ssum = 0.f;
            for (int c = 0; c < NCLS; ++c) ssum += proba[c];
            if (ssum == 0.0f) ssum = 1.0f;
            float inv = 1.0f / ssum;
            int bestc = 0; float bestv = proba[0] * inv;
            out[(size_t)N + (size_t)q * NCLS + 0] = bestv;
            for (int c = 1; c < NCLS; ++c) {
                float pv = proba[c] * inv;
                out[(size_t)N + (size_t)q * NCLS + c] = pv;
                if (pv > bestv) { bestv = pv; bestc = c; }
            }
            out[q] = (float)bestc;   // pred as float (argmax, first max wins)
        }
    }
}

extern "C" void kernel_launch(void* const* d_in, const int* in_sizes, int n_in,
                              void* d_out, int out_size, void* d_ws, size_t ws_size,
                              hipStream_t stream) {
    const float* x      = (const float*)d_in[0];
    const float* train  = (const float*)d_in[1];
    const int*   labels = (const int*)d_in[2];
    float* out = (float*)d_out;

    int N = in_sizes[0] / DIM;   // 2048 queries
    int M = in_sizes[1] / DIM;   // 100000 train points

    int grid = (N + QTILE - 1) / QTILE;
    knn_wmma_kernel<<<grid, 128, 0, stream>>>(x, train, labels, out, N, M);
}